// GAT_42288247996685
// MI455X (gfx1250) — compile-verified
//
#include <hip/hip_runtime.h>

// ---------------------------------------------------------------------------
// GAT forward for gfx1250 (MI455X). wave32, WMMA f32 16x16x4 for all GEMMs.
// Large intermediates (xp 51MB, x_local 51MB, al_* 3.2MB) are L2-resident
// (192MB L2), so edge gathers and the 218M scatter-atomics never touch HBM.
// Cross-lane reductions use v_permlane16 (VALU) instead of ds_bpermute.
// ---------------------------------------------------------------------------

typedef __attribute__((ext_vector_type(2))) float v2f;
typedef __attribute__((ext_vector_type(8))) float v8f;

#define XTP 132  // LDS tile row pitch (floats); 132%64==4 dodges bank conflicts

__device__ __forceinline__ float leaky(float x, float s) { return x > 0.f ? x : s * x; }

// permlane16 butterfly: sel patterns encode lane^1 / ^2 / ^4 / ^8 within each
// 16-lane half; permlanex16 identity handles the cross-half (^16) exchange.
__device__ __forceinline__ float pl16(float v, unsigned lo, unsigned hi) {
  return __uint_as_float(__builtin_amdgcn_permlane16(
      __float_as_uint(v), __float_as_uint(v), lo, hi, false, false));
}
__device__ __forceinline__ float halfsum(float v) {
  v += pl16(v, 0x67452301u, 0xEFCDAB89u);  // xor 1
  v += pl16(v, 0x54761032u, 0xDCFE98BAu);  // xor 2
  v += pl16(v, 0x32107654u, 0xBA98FEDCu);  // xor 4
  v += pl16(v, 0xFEDCBA98u, 0x76543210u);  // xor 8
  return v;  // all lanes of each half hold that half's sum
}
__device__ __forceinline__ float xhalf(float v) {  // value from lane^16
  return __uint_as_float(__builtin_amdgcn_permlanex16(
      __float_as_uint(v), __float_as_uint(v), 0x76543210u, 0xFEDCBA98u, false, false));
}

// Same-wave cross-lane LDS store->load ordering (DS ops are in-order per wave;
// drain DScnt and stop compiler reordering).
__device__ __forceinline__ void lds_fence() {
  asm volatile("s_wait_dscnt 0" ::: "memory");
}

// ---------------------------------------------------------------------------
// Prep kernels.
// k_pack_fcW: pair-packed transpose of fc_W so a WMMA B-fragment (rows k,k+1,
//             one column) is a single contiguous b64:
//             fcp[(k>>1)*256 + j*2 + (k&1)] = fc_W[j][k]
// k_prep_wsd: fused logit matrix, Wsd[d][j] = sum_f W_conv[h][d][f]*a[h][f]
//             (j<8: a_src head j; j>=8: a_dst head j-8), pair-packed the same
//             way: wsd[(d>>1)*32 + j*2 + (d&1)].
// ---------------------------------------------------------------------------
__global__ void k_pack_fcW(const float* __restrict__ W, float* __restrict__ fcp) {
  const int i = blockIdx.x * blockDim.x + threadIdx.x;
  if (i < 128 * 128) {
    const int j = i >> 7, k = i & 127;
    fcp[(k >> 1) * 256 + j * 2 + (k & 1)] = W[i];
  }
}

__global__ void k_prep_wsd(const float* __restrict__ Wc, const float* __restrict__ a_src,
                           const float* __restrict__ a_dst, float* __restrict__ wsd) {
  const int i = threadIdx.x;  // 256 threads, 1 block
  if (i < 256) {
    const int d = i >> 4, j = i & 15;
    const int h = j & 7;
    const float* wrow = Wc + h * 256 + d * 16;
    const float* av = (j < 8 ? a_src : a_dst) + h * 16;
    float s = 0.f;
#pragma unroll
    for (int f = 0; f < 16; ++f) s += wrow[f] * av[f];
    wsd[(d >> 1) * 32 + j * 2 + (d & 1)] = s;
  }
}

// ---------------------------------------------------------------------------
// Kernel 1: per 16-node tile (one wave):
//   xp   = x @ W            (8 heads x 4 WMMA f32 16x16x4)
//   al   = x @ [Ws|Wd]      (4 WMMA; cols 0-7 = al_src, 8-15 = al_dst)
// A frag (16x4 f32): lane<16 -> row=lane%16, k={k0,k0+1}; lane>=16 -> +2.
// C/D: VGPR r = rows r / r+8 per half, lane%16 = column.
// ---------------------------------------------------------------------------
__global__ __launch_bounds__(128) void k1_node_transform(
    const float* __restrict__ x, const float* __restrict__ Wc,
    const float* __restrict__ wsd, float* __restrict__ xp,
    float* __restrict__ als, float* __restrict__ ald, int n) {
  __shared__ float wcp_s[2048];  // W_conv pair-packed per head
  __shared__ float wsd_s[256];
  for (int i = threadIdx.x; i < 2048; i += 128) {
    const int h = i >> 8, d = (i >> 4) & 15, f = i & 15;
    wcp_s[h * 256 + (d >> 1) * 32 + f * 2 + (d & 1)] = Wc[i];
  }
  for (int i = threadIdx.x; i < 256; i += 128) wsd_s[i] = wsd[i];
  __syncthreads();

  const int lane = threadIdx.x & 31;
  const int wave = threadIdx.x >> 5;
  const int gw = blockIdx.x * 4 + wave;
  const int nw = gridDim.x * 4;
  const int ntiles = (n + 15) >> 4;
  const int nidx = lane & 15;
  const int hi = lane >> 4;
  const int kk = hi * 2;
  const int rowoff = hi * 8;
  float* const alp = (nidx < 8) ? als : ald;  // lane-selected logit target
  const int hcol = nidx & 7;

  for (int tile = gw; tile < ntiles; tile += nw) {
    const int base = tile << 4;
    const int arow = base + nidx;
    const bool va = arow < n;
    const float* xr = x + (size_t)(va ? arow : 0) * 16;
    v2f a[4];
#pragma unroll
    for (int kc = 0; kc < 4; ++kc) {
      const v2f t = *(const v2f*)(xr + kc * 4 + kk);
      a[kc].x = va ? t.x : 0.f;
      a[kc].y = va ? t.y : 0.f;
    }
    // attention logits: one 16x16 tile, B = [Ws | Wd]
    v8f cal = {};
#pragma unroll
    for (int kc = 0; kc < 4; ++kc) {
      const int d0 = kc * 4 + kk;
      const v2f b = *(const v2f*)(wsd_s + (d0 >> 1) * 32 + nidx * 2);
      cal = __builtin_amdgcn_wmma_f32_16x16x4_f32(false, a[kc], false, b,
                                                  (short)0, cal, false, false);
    }
    // per-head features
    v8f ch[8];
#pragma unroll
    for (int h = 0; h < 8; ++h) {
      v8f c = {};
#pragma unroll
      for (int kc = 0; kc < 4; ++kc) {
        const int d0 = kc * 4 + kk;
        const v2f b = *(const v2f*)(wcp_s + h * 256 + (d0 >> 1) * 32 + nidx * 2);
        c = __builtin_amdgcn_wmma_f32_16x16x4_f32(false, a[kc], false, b,
                                                  (short)0, c, false, false);
      }
      ch[h] = c;
    }
    if (base + 16 <= n) {  // full tile: straight-line stores
#pragma unroll
      for (int r = 0; r < 8; ++r) {
        const size_t row = (size_t)(base + r + rowoff);
#pragma unroll
        for (int h = 0; h < 8; ++h) xp[row * 128 + h * 16 + nidx] = ch[h][r];
        alp[row * 8 + hcol] = cal[r];
      }
    } else {
#pragma unroll
      for (int r = 0; r < 8; ++r) {
        const int row = base + r + rowoff;
        if (row < n) {
#pragma unroll
          for (int h = 0; h < 8; ++h) xp[(size_t)row * 128 + h * 16 + nidx] = ch[h][r];
          alp[(size_t)row * 8 + hcol] = cal[r];
        }
      }
    }
  }
}

// ---------------------------------------------------------------------------
// Init kernels (b128 vectorized; C=128 is a multiple of 4)
// ---------------------------------------------------------------------------
__global__ void k2_init_xlocal(float4* __restrict__ xl4,
                               const float4* __restrict__ bc4, int total4) {
  const int i = blockIdx.x * blockDim.x + threadIdx.x;
  if (i < total4) xl4[i] = bc4[i & 31];  // seed with bias; atomics accumulate
}

__global__ void k2_zero(float* __restrict__ denom, int nd, float* __restrict__ xg) {
  const int i = blockIdx.x * blockDim.x + threadIdx.x;
  if (i < nd) denom[i] = 0.f;
  if (i < 128) xg[i] = 0.f;
}

// ---------------------------------------------------------------------------
// Kernel 3: softmax denominators per (dst, head). Softmax is shift-invariant,
// so the segment-max pass is skipped (mathematically identical; logits O(1)).
// ---------------------------------------------------------------------------
__global__ void k3_denominator(const int* __restrict__ ei, int E, int n,
                               const float* __restrict__ als,
                               const float* __restrict__ ald,
                               float* __restrict__ denom) {
  const int e = blockIdx.x * blockDim.x + threadIdx.x;
  if (e >= E + n) return;
  int s, d;
  if (e < E) {
    s = ei[e]; d = ei[E + e];
    s = s < 0 ? 0 : (s >= n ? n - 1 : s);
    d = d < 0 ? 0 : (d >= n ? n - 1 : d);
  } else {
    s = d = e - E;  // self loop
  }
  const float4 s0 = *(const float4*)(als + (size_t)s * 8);
  const float4 s1 = *(const float4*)(als + (size_t)s * 8 + 4);
  const float4 d0 = *(const float4*)(ald + (size_t)d * 8);
  const float4 d1 = *(const float4*)(ald + (size_t)d * 8 + 4);
  const float v[8] = {s0.x + d0.x, s0.y + d0.y, s0.z + d0.z, s0.w + d0.w,
                      s1.x + d1.x, s1.y + d1.y, s1.z + d1.z, s1.w + d1.w};
  float* drow = denom + (size_t)d * 8;
#pragma unroll
  for (int h = 0; h < 8; ++h) atomicAdd(drow + h, __expf(leaky(v[h], 0.2f)));
}

// ---------------------------------------------------------------------------
// Kernel 4: message scatter. One thread per (edge, head): alpha * xp[src] row
// (64B contiguous) atomically added into x_local[dst]. Atomics land in L2.
// ---------------------------------------------------------------------------
__global__ void k4_scatter(const int* __restrict__ ei, int E, int n,
                           const float* __restrict__ als,
                           const float* __restrict__ ald,
                           const float* __restrict__ denom,
                           const float* __restrict__ xp, float* __restrict__ xl) {
  const int i = blockIdx.x * blockDim.x + threadIdx.x;
  if (i >= (E + n) * 8) return;
  const int e = i >> 3, h = i & 7;
  int s, d;
  if (e < E) {
    s = ei[e]; d = ei[E + e];
    s = s < 0 ? 0 : (s >= n ? n - 1 : s);
    d = d < 0 ? 0 : (d >= n ? n - 1 : d);
  } else {
    s = d = e - E;
  }
  const float t = als[(size_t)s * 8 + h] + ald[(size_t)d * 8 + h];
  const float alpha = __expf(leaky(t, 0.2f)) / denom[(size_t)d * 8 + h];
  const float4* srcv = (const float4*)(xp + (size_t)s * 128 + h * 16);
  float* dstp = xl + (size_t)d * 128 + h * 16;
#pragma unroll
  for (int q = 0; q < 4; ++q) {
    const float4 v = srcv[q];
    atomicAdd(dstp + q * 4 + 0, v.x * alpha);
    atomicAdd(dstp + q * 4 + 1, v.y * alpha);
    atomicAdd(dstp + q * 4 + 2, v.z * alpha);
    atomicAdd(dstp + q * 4 + 3, v.w * alpha);
  }
}

// ---------------------------------------------------------------------------
// 16x128 tile GEMM: acc = X(16x128, LDS) @ fc_W^T + fc_b.
// B frags come from pair-packed fcp (one b64, 16 lanes = 128B contiguous).
// 32 K-chunks x 8 col-tiles = 256 WMMA f32 16x16x4.
// ---------------------------------------------------------------------------
__device__ __forceinline__ void gemm_tile16x128(const float* xt,
                                                const float* __restrict__ fcp,
                                                const float* __restrict__ fc_b,
                                                v8f acc[8], int kk, int nidx) {
#pragma unroll
  for (int j = 0; j < 8; ++j) { v8f z = {}; acc[j] = z; }
  for (int k0 = 0; k0 < 128; k0 += 4) {
    const v2f a = *(const v2f*)(xt + nidx * XTP + k0 + kk);
    const float* bbase = fcp + ((k0 + kk) >> 1) * 256;
#pragma unroll
    for (int j = 0; j < 8; ++j) {
      const v2f b = *(const v2f*)(bbase + (j * 16 + nidx) * 2);
      acc[j] = __builtin_amdgcn_wmma_f32_16x16x4_f32(false, a, false, b,
                                                     (short)0, acc[j], false, false);
    }
  }
#pragma unroll
  for (int j = 0; j < 8; ++j) {
    const float bb = fc_b[j * 16 + nidx];
#pragma unroll
    for (int r = 0; r < 8; ++r) acc[j][r] += bb;
  }
}

// ---------------------------------------------------------------------------
// Kernel 5: fused fc/att/fc/layernorm/l2norm per 16-node tile per wave.
// ---------------------------------------------------------------------------
__global__ __launch_bounds__(128) void k5_fc_path(
    const float* __restrict__ xloc, const float* __restrict__ fcp,
    const float* __restrict__ fc_b, const float* __restrict__ ln_g,
    const float* __restrict__ ln_b, float* __restrict__ out,
    float* __restrict__ xg_accum, int n) {
  __shared__ float xt_all[4][16 * XTP];
  const int lane = threadIdx.x & 31;
  const int wave = threadIdx.x >> 5;
  float* xt = xt_all[wave];
  const int gw = blockIdx.x * 4 + wave;
  const int nw = gridDim.x * 4;
  const int ntiles = (n + 15) >> 4;
  const int nidx = lane & 15;
  const int hi = lane >> 4;
  const int kk = hi * 2;
  const int rowoff = hi * 8;

  float xg_loc[8];
#pragma unroll
  for (int j = 0; j < 8; ++j) xg_loc[j] = 0.f;

  for (int tile = gw; tile < ntiles; tile += nw) {
    const int base = tile << 4;
    const bool full = (base + 16 <= n);
    // stage tile as b128: tile is contiguous in global; XTP*4=528 keeps 16B align
    if (full) {
      const float4* src4 = (const float4*)(xloc + (size_t)base * 128);
      for (int i = lane; i < 512; i += 32) {
        const int r = i >> 5, c4 = i & 31;
        *(float4*)(xt + r * XTP + c4 * 4) = src4[i];
      }
    } else {
      const float4 z4 = make_float4(0.f, 0.f, 0.f, 0.f);
      for (int i = lane; i < 512; i += 32) {
        const int r = i >> 5, c4 = i & 31;
        *(float4*)(xt + r * XTP + c4 * 4) =
            (base + r < n) ? ((const float4*)xloc)[(size_t)(base + r) * 32 + c4] : z4;
      }
    }
    lds_fence();

    v8f acc[8];
    gemm_tile16x128(xt, fcp, fc_b, acc, kk, nidx);  // y1 = X@fc_W^T + fc_b

    // att = softmax(leaky_relu(y1, 0.01), axis=-1) (shift-invariant form)
    float rs[8];
#pragma unroll
    for (int r = 0; r < 8; ++r) rs[r] = 0.f;
#pragma unroll
    for (int j = 0; j < 8; ++j)
#pragma unroll
      for (int r = 0; r < 8; ++r) {
        const float ev = __expf(leaky(acc[j][r], 0.01f));
        acc[j][r] = ev;
        rs[r] += ev;
      }
#pragma unroll
    for (int r = 0; r < 8; ++r) rs[r] = halfsum(rs[r]);

    // xl0 = leaky_relu(X * att, 0.2) -> re-layout via LDS for GEMM #2 A-frags
#pragma unroll
    for (int j = 0; j < 8; ++j) {
      const int col = j * 16 + nidx;
#pragma unroll
      for (int r = 0; r < 8; ++r) {
        const int row = r + rowoff;
        const float v = leaky(xt[row * XTP + col] * (acc[j][r] / rs[r]), 0.2f);
        xt[row * XTP + col] = v;
      }
    }
    lds_fence();

    gemm_tile16x128(xt, fcp, fc_b, acc, kk, nidx);  // y2 = xl0@fc_W^T + fc_b

    // layer norm (eps 1e-5): one pass, var = E[x^2] - E[x]^2
    float mu[8], iv[8];
#pragma unroll
    for (int r = 0; r < 8; ++r) {
      float s1 = 0.f, s2 = 0.f;
#pragma unroll
      for (int j = 0; j < 8; ++j) { const float v = acc[j][r]; s1 += v; s2 += v * v; }
      s1 = halfsum(s1);
      s2 = halfsum(s2);
      mu[r] = s1 * (1.f / 128.f);
      iv[r] = rsqrtf(fmaxf(s2 * (1.f / 128.f) - mu[r] * mu[r], 0.f) + 1e-5f);
    }
#pragma unroll
    for (int j = 0; j < 8; ++j) {
      const int col = j * 16 + nidx;
      const float g = ln_g[col], bt = ln_b[col];
#pragma unroll
      for (int r = 0; r < 8; ++r) acc[j][r] = (acc[j][r] - mu[r]) * iv[r] * g + bt;
    }

    // L2 row normalize (eps 1e-12)
#pragma unroll
    for (int r = 0; r < 8; ++r) {
      float s = 0.f;
#pragma unroll
      for (int j = 0; j < 8; ++j) s += acc[j][r] * acc[j][r];
      const float inv = 1.f / fmaxf(sqrtf(halfsum(s)), 1e-12f);
#pragma unroll
      for (int j = 0; j < 8; ++j) acc[j][r] *= inv;
    }

    // store xl, accumulate column sums for the global mean
    if (full) {
#pragma unroll
      for (int j = 0; j < 8; ++j) {
        const int col = j * 16 + nidx;
#pragma unroll
        for (int r = 0; r < 8; ++r) {
          out[(size_t)(base + r + rowoff) * 128 + col] = acc[j][r];
          xg_loc[j] += acc[j][r];
        }
      }
    } else {
#pragma unroll
      for (int j = 0; j < 8; ++j) {
        const int col = j * 16 + nidx;
#pragma unroll
        for (int r = 0; r < 8; ++r) {
          const int row = base + r + rowoff;
          if (row < n) {
            out[(size_t)row * 128 + col] = acc[j][r];
            xg_loc[j] += acc[j][r];
          }
        }
      }
    }
  }

  // flush per-wave column sums (one atomic per column per wave)
#pragma unroll
  for (int j = 0; j < 8; ++j) {
    const float t = xg_loc[j] + xhalf(xg_loc[j]);
    if (lane < 16) atomicAdd(&xg_accum[j * 16 + nidx], t);
  }
}

// ---------------------------------------------------------------------------
// Kernel 6: ga = softmax(relu(mean(xl) @ gfc_W^T + gfc_b))   (single block)
// ---------------------------------------------------------------------------
__global__ __launch_bounds__(128) void k6_global_attn(
    const float* __restrict__ xg_accum, const float* __restrict__ gW,
    const float* __restrict__ gb, float* __restrict__ ga, int n) {
  __shared__ float xg[128];
  __shared__ float tv[128];
  const int c = threadIdx.x;
  xg[c] = xg_accum[c] / (float)n;
  __syncthreads();
  float t = gb[c];
  for (int k = 0; k < 128; ++k) t += xg[k] * gW[c * 128 + k];
  t = fmaxf(t, 0.f);
  tv[c] = t;
  __syncthreads();
  float mx = -1e30f;
  for (int k = 0; k < 128; ++k) mx = fmaxf(mx, tv[k]);
  const float e = __expf(t - mx);
  __syncthreads();
  tv[c] = e;
  __syncthreads();
  float s = 0.f;
  for (int k = 0; k < 128; ++k) s += tv[k];
  ga[c] = e / s;
}

__global__ void k7_scale(float4* __restrict__ out4, const float4* __restrict__ ga4,
                         int total4) {
  const int i = blockIdx.x * blockDim.x + threadIdx.x;
  if (i < total4) {
    const float4 g = ga4[i & 31];
    float4 v = out4[i];
    v.x *= g.x; v.y *= g.y; v.z *= g.z; v.w *= g.w;
    out4[i] = v;
  }
}

// ---------------------------------------------------------------------------
extern "C" void kernel_launch(void* const* d_in, const int* in_sizes, int n_in,
                              void* d_out, int out_size, void* d_ws, size_t ws_size,
                              hipStream_t stream) {
  (void)n_in; (void)out_size; (void)ws_size;
  const float* x      = (const float*)d_in[0];
  const int*   ei     = (const int*)d_in[1];
  const float* Wc     = (const float*)d_in[2];
  const float* a_src  = (const float*)d_in[3];
  const float* a_dst  = (const float*)d_in[4];
  const float* b_conv = (const float*)d_in[5];
  const float* fc_W   = (const float*)d_in[6];
  const float* fc_b   = (const float*)d_in[7];
  const float* ln_g   = (const float*)d_in[8];
  const float* ln_b   = (const float*)d_in[9];
  const float* gfc_W  = (const float*)d_in[10];
  const float* gfc_b  = (const float*)d_in[11];
  const int n = in_sizes[0] / 16;
  const int E = in_sizes[1] / 2;
  float* out = (float*)d_out;

  char* ws = (char*)d_ws;
  auto rnd = [](size_t b) { return (b + 255) & ~(size_t)255; };
  size_t o = 0;
  float* xp    = (float*)(ws + o); o += rnd((size_t)n * 128 * 4);
  float* xloc  = (float*)(ws + o); o += rnd((size_t)n * 128 * 4);
  float* als   = (float*)(ws + o); o += rnd((size_t)n * 8 * 4);
  float* ald   = (float*)(ws + o); o += rnd((size_t)n * 8 * 4);
  float* denom = (float*)(ws + o); o += rnd((size_t)n * 8 * 4);
  float* fcp   = (float*)(ws + o); o += rnd(128 * 128 * 4);
  float* wsd   = (float*)(ws + o); o += rnd(256 * 4);
  float* xg    = (float*)(ws + o); o += rnd(128 * 4);
  float* ga    = (float*)(ws + o); o += rnd(128 * 4);

  const int ntiles = (n + 15) / 16;
  const int tblocks = (ntiles + 3) / 4;  // 4 waves (tiles) per 128-thread block
  const int tot_e = E + n;
  const int total4 = n * 32;             // N*128 floats as float4

  k_pack_fcW<<<(128 * 128 + 255) / 256, 256, 0, stream>>>(fc_W, fcp);
  k_prep_wsd<<<1, 256, 0, stream>>>(Wc, a_src, a_dst, wsd);
  k1_node_transform<<<tblocks, 128, 0, stream>>>(x, Wc, wsd, xp, als, ald, n);
  k2_init_xlocal<<<(total4 + 255) / 256, 256, 0, stream>>>(
      (float4*)xloc, (const float4*)b_conv, total4);
  k2_zero<<<(n * 8 + 255) / 256, 256, 0, stream>>>(denom, n * 8, xg);
  k3_denominator<<<(tot_e + 255) / 256, 256, 0, stream>>>(ei, E, n, als, ald, denom);
  k4_scatter<<<(tot_e * 8 + 255) / 256, 256, 0, stream>>>(ei, E, n, als, ald, denom, xp, xloc);
  k5_fc_path<<<tblocks, 128, 0, stream>>>(xloc, fcp, fc_b, ln_g, ln_b, out, xg, n);
  k6_global_attn<<<1, 128, 0, stream>>>(xg, gfc_W, gfc_b, ga, n);
  k7_scale<<<(total4 + 255) / 256, 256, 0, stream>>>(
      (float4*)out, (const float4*)ga, total4);
}